// GNN_Block_sparse_48275432407773
// MI455X (gfx1250) — compile-verified
//
#include <hip/hip_runtime.h>
#include <stdint.h>

#define D 128  // feature width for x, x1, x2 and all GEMM N dims

typedef __attribute__((ext_vector_type(2))) float v2f;
typedef __attribute__((ext_vector_type(8))) float v8f;

// ---------------------------------------------------------------------------
// Degree / normalization:  deg[i] = 1 (self loop) + #incoming edges at i
// then dinv[i] = rsqrt(deg[i])  (deg >= 1 always, so no zero guard needed)
// ---------------------------------------------------------------------------
__global__ void deg_init_kernel(float* __restrict__ deg, int n) {
  int i = blockIdx.x * blockDim.x + threadIdx.x;
  if (i < n) deg[i] = 1.0f;  // self loop contribution
}

__global__ void deg_edges_kernel(const int* __restrict__ col,
                                 float* __restrict__ deg, int e) {
  int i = blockIdx.x * blockDim.x + threadIdx.x;
  if (i < e) atomicAdd(&deg[col[i]], 1.0f);
}

__global__ void dinv_kernel(float* __restrict__ deg, int n) {
  int i = blockIdx.x * blockDim.x + threadIdx.x;
  if (i < n) deg[i] = rsqrtf(deg[i]);
}

// ---------------------------------------------------------------------------
// GEMM: C[M x 128] = A[M x K] @ W[K x 128] (+ optional bias)
// A is given as two halves (for the concat GEMM): k <  128 reads A0[m][k],
//                                                 k >= 128 reads A1[m][k-128].
// One wave32 per 16x16 output tile, K stepped by 4 with
// V_WMMA_F32_16X16X4_F32.
//   A frag (16x4 f32): lanes 0-15 -> M=lane,    VGPR0=K+0, VGPR1=K+1
//                      lanes 16-31-> M=lane-16, VGPR0=K+2, VGPR1=K+3
//   B frag (4x16 f32): lanes 0-15 -> N=lane,    VGPR0=K+0 row, VGPR1=K+1 row
//                      lanes 16-31-> N=lane-16, VGPR0=K+2 row, VGPR1=K+3 row
//   C/D (16x16 f32, 8 VGPRs): VGPR r: lanes 0-15 -> (M=r, N=lane),
//                                      lanes 16-31 -> (M=r+8, N=lane-16)
// Requires M % 16 == 0 (true here: 50000 = 3125*16).
// ---------------------------------------------------------------------------
__global__ void gemm_wmma_kernel(const float* __restrict__ A0,
                                 const float* __restrict__ A1,
                                 const float* __restrict__ W,
                                 const float* __restrict__ bias,
                                 float* __restrict__ C, int M, int K) {
  const int tiles_n = D / 16;  // 8
  int wave_id = (blockIdx.x * blockDim.x + threadIdx.x) >> 5;
  int lane = threadIdx.x & 31;
  int tm = wave_id / tiles_n;
  int tn = wave_id % tiles_n;
  if (tm >= M / 16) return;  // wave-uniform: EXEC stays all-1s for WMMA

  const int row0 = tm * 16;
  const int col0 = tn * 16;
  const int half = lane >> 4;  // 0: lanes 0-15, 1: lanes 16-31
  const int l15 = lane & 15;
  const int n = col0 + l15;

  v8f acc = {};
  for (int k = 0; k < K; k += 4) {
    const float* Asrc = (k < D) ? A0 : A1;
    const int kk = (k < D) ? k : (k - D);
    // A fragment: two consecutive K values for this lane's row
    const float* arow = Asrc + (size_t)(row0 + l15) * D + kk + half * 2;
    v2f a;
    a.x = arow[0];
    a.y = arow[1];
    // B fragment: two consecutive K rows at this lane's column
    const float* bptr = W + (size_t)(k + half * 2) * D + n;
    v2f b;
    b.x = bptr[0];
    b.y = bptr[D];
#if defined(__AMDGCN__) && __has_builtin(__builtin_amdgcn_wmma_f32_16x16x4_f32)
    acc = __builtin_amdgcn_wmma_f32_16x16x4_f32(
        /*neg_a=*/false, a, /*neg_b=*/false, b,
        /*c_mod=*/(short)0, acc, /*reuse_a=*/false, /*reuse_b=*/false);
#else
    // Scalar fallback matching the same per-lane accumulator layout.
#pragma unroll
    for (int r = 0; r < 8; ++r) {
      const int m = row0 + r + half * 8;
      float s = acc[r];
#pragma unroll
      for (int kq = 0; kq < 4; ++kq) {
        const float av = Asrc[(size_t)m * D + kk + kq];
        const float bv = W[(size_t)(k + kq) * D + n];
        s = fmaf(av, bv, s);
      }
      acc[r] = s;
    }
#endif
  }

  const float bv = bias ? bias[n] : 0.0f;
#pragma unroll
  for (int r = 0; r < 8; ++r) {
    const int m = row0 + r + half * 8;
    C[(size_t)m * D + n] = acc[r] + bv;
  }
}

// ---------------------------------------------------------------------------
// Aggregation: agg[i] = dinv[i]^2 * h[i]  (self loop term), then
//              agg[c] += dinv[r]*dinv[c] * h[r] over edges (atomic scatter),
// then x = relu(agg + b).
// ---------------------------------------------------------------------------
__global__ void self_loop_kernel(const float* __restrict__ h,
                                 const float* __restrict__ dinv,
                                 float* __restrict__ agg, int n) {
  int idx = blockIdx.x * blockDim.x + threadIdx.x;
  if (idx < n * D) {
    int i = idx >> 7;  // D == 128
    float d = dinv[i];
    agg[idx] = h[idx] * d * d;
  }
}

__global__ void edge_msg_kernel(const int* __restrict__ row,
                                const int* __restrict__ col,
                                const float* __restrict__ h,
                                const float* __restrict__ dinv,
                                float* __restrict__ agg, int e) {
  // 256 threads = 2 edges/block; lanes within a wave share one edge, so the
  // row/col/dinv loads are wave-uniform and the h gather is fully coalesced.
  int eid = blockIdx.x * 2 + (threadIdx.x >> 7);
  int f = threadIdx.x & (D - 1);
  if (eid < e) {
    int r = row[eid];
    int c = col[eid];
    float norm = dinv[r] * dinv[c];
    atomicAdd(&agg[(size_t)c * D + f], h[(size_t)r * D + f] * norm);
  }
}

__global__ void bias_relu_kernel(float* __restrict__ agg,
                                 const float* __restrict__ b, int n) {
  int idx = blockIdx.x * blockDim.x + threadIdx.x;
  if (idx < n * D) {
    float v = agg[idx] + b[idx & (D - 1)];
    agg[idx] = fmaxf(v, 0.0f);
  }
}

// ---------------------------------------------------------------------------
// Launch
// ---------------------------------------------------------------------------
static inline size_t align_up(size_t x, size_t a) { return (x + a - 1) & ~(a - 1); }

extern "C" void kernel_launch(void* const* d_in, const int* in_sizes, int n_in,
                              void* d_out, int out_size, void* d_ws, size_t ws_size,
                              hipStream_t stream) {
  (void)n_in; (void)out_size; (void)ws_size;

  const float* x  = (const float*)d_in[0];
  const int*   ei = (const int*)d_in[1];   // edge_index, [2, E] flat
  const float* W1 = (const float*)d_in[2];
  const float* b1 = (const float*)d_in[3];
  const float* W2 = (const float*)d_in[4];
  const float* b2 = (const float*)d_in[5];
  const float* Wl = (const float*)d_in[6];  // [256, 128] row-major
  const float* bl = (const float*)d_in[7];

  const int N = in_sizes[0] / D;   // 50000
  const int E = in_sizes[1] / 2;   // 600000
  const int* row = ei;
  const int* col = ei + E;

  // Workspace carve-up
  char* base = (char*)d_ws;
  size_t off = 0;
  float* dinv = (float*)(base + off); off += align_up((size_t)N * 4, 256);
  float* h    = (float*)(base + off); off += align_up((size_t)N * D * 4, 256);
  float* x1   = (float*)(base + off); off += align_up((size_t)N * D * 4, 256);
  float* x2   = (float*)(base + off); off += align_up((size_t)N * D * 4, 256);

  const int THREADS = 256;
  dim3 blk(THREADS);
  dim3 grid_n((N + THREADS - 1) / THREADS);
  dim3 grid_e((E + THREADS - 1) / THREADS);
  dim3 grid_nf(((size_t)N * D + THREADS - 1) / THREADS);
  dim3 grid_edge((E + 1) / 2);
  // GEMM: (N/16)*8 tiles, 8 waves (tiles) per 256-thread block
  dim3 grid_gemm(((N / 16) * (D / 16) + 7) / 8);

  // 1) degree -> dinv
  deg_init_kernel<<<grid_n, blk, 0, stream>>>(dinv, N);
  deg_edges_kernel<<<grid_e, blk, 0, stream>>>(col, dinv, E);
  dinv_kernel<<<grid_n, blk, 0, stream>>>(dinv, N);

  // 2) layer 1: h = x @ W1 ; aggregate ; +b1 ; relu -> x1
  gemm_wmma_kernel<<<grid_gemm, blk, 0, stream>>>(x, nullptr, W1, nullptr, h, N, D);
  self_loop_kernel<<<grid_nf, blk, 0, stream>>>(h, dinv, x1, N);
  edge_msg_kernel<<<grid_edge, blk, 0, stream>>>(row, col, h, dinv, x1, E);
  bias_relu_kernel<<<grid_nf, blk, 0, stream>>>(x1, b1, N);

  // 3) layer 2: h = x1 @ W2 ; aggregate ; +b2 ; relu -> x2
  gemm_wmma_kernel<<<grid_gemm, blk, 0, stream>>>(x1, nullptr, W2, nullptr, h, N, D);
  self_loop_kernel<<<grid_nf, blk, 0, stream>>>(h, dinv, x2, N);
  edge_msg_kernel<<<grid_edge, blk, 0, stream>>>(row, col, h, dinv, x2, E);
  bias_relu_kernel<<<grid_nf, blk, 0, stream>>>(x2, b2, N);

  // 4) head: out = concat([x1, x2]) @ Wl + bl  (K = 256, no materialized concat)
  gemm_wmma_kernel<<<grid_gemm, blk, 0, stream>>>(x1, x2, Wl, bl, (float*)d_out, N, 2 * D);
}